// Op_78915729096709
// MI455X (gfx1250) — compile-verified
//
#include <hip/hip_runtime.h>
#include <stdint.h>

// Problem constants (from the reference): B,TAPE,N,F = 128,100000,50000,32
constexpr int kB    = 128;
constexpr int kTAPE = 100000;
constexpr int kN    = 50000;
constexpr int kF    = 32;
constexpr int kNT   = 128;   // neurons per block in the main kernel
constexpr int kBLK  = 256;   // 8 waves (wave32)

// v4i exactly as the builtin's parameter type was printed by the compiler:
//   '__attribute__((__vector_size__(4 * sizeof(int)))) int *'
typedef int v4i __attribute__((__vector_size__(4 * sizeof(int))));

#if defined(__HIP_DEVICE_COMPILE__)
#  if __has_builtin(__builtin_amdgcn_global_load_async_to_lds_b128)
#    define HAVE_ASYNC_LDS 1
#  else
#    define HAVE_ASYNC_LDS 0
#    warning "device pass: __builtin_amdgcn_global_load_async_to_lds_b128 NOT found; synchronous staging fallback in use"
#  endif
#  if __has_builtin(__builtin_amdgcn_s_wait_asynccnt)
#    define WAIT_ASYNCCNT0() __builtin_amdgcn_s_wait_asynccnt(0)
#  else
#    define WAIT_ASYNCCNT0() asm volatile("s_wait_asynccnt 0" ::: "memory")
#  endif
#else
#  define HAVE_ASYNC_LDS 0
#  define WAIT_ASYNCCNT0()
#endif

// ---------------- Kernel 1: base copy tape -> out (b128 path) --------------
__global__ __launch_bounds__(kBLK) void copy_b128(const float4* __restrict__ src,
                                                  float4* __restrict__ dst, int n4) {
  const int i = blockIdx.x * kBLK + threadIdx.x;
  if (i < n4) dst[i] = src[i];
}

// ---------------- Kernel 2: transpose tape (B x TAPE) -> tapeT (TAPE x B) --
// 32x32 LDS tiles, padded stride 33 -> conflict-free. TAPE%32==0, B%32==0.
__global__ __launch_bounds__(kBLK) void transpose_tape(const float* __restrict__ tape,
                                                       float* __restrict__ tapeT) {
  __shared__ float t[32][33];
  const int c0 = blockIdx.x * 32;   // tape column base
  const int b0 = blockIdx.y * 32;   // batch row base
  const int lx = threadIdx.x;       // 0..31
  const int ly = threadIdx.y;       // 0..7
  #pragma unroll
  for (int j = ly; j < 32; j += 8)
    t[j][lx] = tape[(size_t)(b0 + j) * kTAPE + (c0 + lx)];
  __syncthreads();
  #pragma unroll
  for (int j = ly; j < 32; j += 8)
    tapeT[(size_t)(c0 + j) * kB + (b0 + lx)] = t[lx][j];
}

// ---------------- Kernel 3: main gather-dot-activate -----------------------
// One block = 128 neurons. Indices+weights staged into LDS with CDNA5 async
// global->LDS b128 copies (ASYNCcnt). Each wave handles one neuron per pass:
// lane owns 4 batch rows (float4), the 32 (idx,w) pairs live one-per-lane and
// are broadcast per f-step via __shfl (v_readlane on wave32). Every gather is
// a fully coalesced 512B global_load_b128 of tapeT[c][0:128], reused by all B.
__global__ __launch_bounds__(kBLK) void mlp_main(const float* __restrict__ tapeT,
                                                 const float* __restrict__ weights,
                                                 const float* __restrict__ bias,
                                                 const int*   __restrict__ iidx,
                                                 const int*   __restrict__ act,
                                                 float*       __restrict__ xT) {
  __shared__ __align__(16) int   s_idx[kNT * kF];   // [n][f], 16 KB
  __shared__ __align__(16) float s_w  [kNT * kF];   // [n][f], 16 KB

  const int tid  = threadIdx.x;
  const int lane = tid & 31;
  const int wave = tid >> 5;                        // 0..7
  const int n0   = blockIdx.x * kNT;
  const int cnt  = min(kNT, kN - n0);               // 128 or 80 (last block)

  // ---- stage index+weight tile: global -> LDS (async on CDNA5) ----
  {
    const int*   gi = iidx    + n0 * kF;
    const float* gw = weights + n0 * kF;
    const int e4cnt = (cnt * kF) >> 2;              // 16B-granule count
    for (int e = tid; e < e4cnt; e += kBLK) {
#if HAVE_ASYNC_LDS
      __builtin_amdgcn_global_load_async_to_lds_b128((v4i*)(gi + 4 * e),
                                                     (v4i*)(&s_idx[4 * e]), 0, 0);
      __builtin_amdgcn_global_load_async_to_lds_b128((v4i*)(gw + 4 * e),
                                                     (v4i*)(&s_w[4 * e]), 0, 0);
#else
      reinterpret_cast<int4*>(s_idx)[e]  = reinterpret_cast<const int4*>(gi)[e];
      reinterpret_cast<float4*>(s_w)[e]  = reinterpret_cast<const float4*>(gw)[e];
#endif
    }
  }
  WAIT_ASYNCCNT0();
  __syncthreads();

  // ---- compute: wave w handles neurons n0 + w, w+8, ... ----
  for (int nn = wave; nn < cnt; nn += 8) {
    const int   n  = n0 + nn;
    const int   cl = s_idx[nn * kF + lane];         // lane-consecutive: no bank conflicts
    const float wl = s_w  [nn * kF + lane];
    const float bv = bias[n];                        // wave-uniform broadcast load
    const int   at = act[n];

    float4 acc; acc.x = bv; acc.y = bv; acc.z = bv; acc.w = bv;
    #pragma unroll
    for (int f = 0; f < kF; ++f) {
      const int   c  = __shfl(cl, f, 32);            // broadcast idx[n][f]
      const float wv = __shfl(wl, f, 32);            // broadcast w[n][f]
      const float4 t = *(const float4*)(tapeT + (size_t)c * kB + lane * 4);
      acc.x = fmaf(t.x, wv, acc.x);
      acc.y = fmaf(t.y, wv, acc.y);
      acc.z = fmaf(t.z, wv, acc.z);
      acc.w = fmaf(t.w, wv, acc.w);
    }
    float4 r;
    if (at == 0) {
      r.x = fmaxf(acc.x, 0.0f); r.y = fmaxf(acc.y, 0.0f);
      r.z = fmaxf(acc.z, 0.0f); r.w = fmaxf(acc.w, 0.0f);
    } else {
      r.x = tanhf(acc.x); r.y = tanhf(acc.y);
      r.z = tanhf(acc.z); r.w = tanhf(acc.w);
    }
    *(float4*)(xT + (size_t)n * kB + lane * 4) = r;  // coalesced b128 store
  }
}

// ---------------- Kernel 4: un-transpose + scatter xT[n][b] -> out[b][oidx[n]]
__global__ __launch_bounds__(kBLK) void scatter_out(const float* __restrict__ xT,
                                                    const int*   __restrict__ oidx,
                                                    float*       __restrict__ out) {
  __shared__ float t[32][33];
  const int n0 = blockIdx.x * 32;
  const int b0 = blockIdx.y * 32;
  const int lx = threadIdx.x;   // 0..31
  const int ly = threadIdx.y;   // 0..7
  #pragma unroll
  for (int j = ly; j < 32; j += 8) {
    const int n = n0 + j;
    if (n < kN) t[j][lx] = xT[(size_t)n * kB + (b0 + lx)];
  }
  __syncthreads();
  const int  n = n0 + lx;
  const bool v = (n < kN);
  const int  on = v ? oidx[n] : 0;
  #pragma unroll
  for (int j = ly; j < 32; j += 8) {
    if (v) out[(size_t)(b0 + j) * kTAPE + on] = t[lx][j];
  }
}

// ---------------- Fallback (workspace too small): direct gather ------------
__global__ __launch_bounds__(kBLK) void fallback_gather(const float* __restrict__ tape,
                                                        const float* __restrict__ weights,
                                                        const float* __restrict__ bias,
                                                        const int*   __restrict__ iidx,
                                                        const int*   __restrict__ oidx,
                                                        const int*   __restrict__ act,
                                                        float*       __restrict__ out) {
  const long long i = (long long)blockIdx.x * kBLK + threadIdx.x;
  if (i >= (long long)kB * kN) return;
  const int n = (int)(i % kN);
  const int b = (int)(i / kN);
  const float* trow = tape + (size_t)b * kTAPE;
  const int*   ir   = iidx + n * kF;
  const float* wr   = weights + n * kF;
  float acc = bias[n];
  #pragma unroll 8
  for (int f = 0; f < kF; ++f) acc = fmaf(trow[ir[f]], wr[f], acc);
  acc = (act[n] == 0) ? fmaxf(acc, 0.0f) : tanhf(acc);
  out[(size_t)b * kTAPE + oidx[n]] = acc;
}

// ---------------- Host launcher --------------------------------------------
extern "C" void kernel_launch(void* const* d_in, const int* in_sizes, int n_in,
                              void* d_out, int out_size, void* d_ws, size_t ws_size,
                              hipStream_t stream) {
  (void)in_sizes; (void)n_in; (void)out_size;
  const float* tape    = (const float*)d_in[0];
  const float* weights = (const float*)d_in[1];
  const float* bias    = (const float*)d_in[2];
  const int*   iidx    = (const int*)d_in[3];
  const int*   oidx    = (const int*)d_in[4];
  const int*   act     = (const int*)d_in[5];
  float*       out     = (float*)d_out;

  // 1) out = tape (covers all columns; scatter overwrites targeted ones)
  const int n4 = kB * kTAPE / 4;
  copy_b128<<<(n4 + kBLK - 1) / kBLK, kBLK, 0, stream>>>(
      (const float4*)tape, (float4*)out, n4);

  const size_t need = (size_t)(kTAPE + kN) * kB * sizeof(float);  // ~76.8 MB
  if (ws_size >= need) {
    float* tapeT = (float*)d_ws;                       // TAPE x B
    float* xT    = tapeT + (size_t)kTAPE * kB;         // N x B
    transpose_tape<<<dim3(kTAPE / 32, kB / 32), dim3(32, 8), 0, stream>>>(tape, tapeT);
    mlp_main<<<(kN + kNT - 1) / kNT, kBLK, 0, stream>>>(tapeT, weights, bias, iidx, act, xT);
    scatter_out<<<dim3((kN + 31) / 32, kB / 32), dim3(32, 8), 0, stream>>>(xT, oidx, out);
  } else {
    const long long total = (long long)kB * kN;
    fallback_gather<<<(int)((total + kBLK - 1) / kBLK), kBLK, 0, stream>>>(
        tape, weights, bias, iidx, oidx, act, out);
  }
}